// Live_Gandalf_Model_Reward_57466662420630
// MI455X (gfx1250) — compile-verified
//
#include <hip/hip_runtime.h>
#include <hip/hip_bf16.h>

typedef float v2f __attribute__((ext_vector_type(2)));
typedef float v8f __attribute__((ext_vector_type(8)));

#define B_SZ   2048
#define F_SZ   512
#define H_SZ   64
#define K_SZ   8
#define D_SZ   128
#define ASR    128
#define MM     768
#define OUT    256
#define PIN    66432            // 128 + 768 + 512*128
#define KC     4992             // 128 + 768 + 512*8
#define TOPK   896              // ASR + MM

// ---------------------------------------------------------------------------
// Wc[0:896, :] = proj_w[:, 0:896]^T
__global__ __launch_bounds__(256) void k_build_wc_top(const float* __restrict__ pw,
                                                      float* __restrict__ wc) {
    int idx = blockIdx.x * blockDim.x + threadIdx.x;
    int n = TOPK * OUT;
    for (; idx < n; idx += gridDim.x * blockDim.x) {
        int r = idx / OUT;     // proj input col 0..895
        int o = idx % OUT;     // output unit
        wc[idx] = pw[(size_t)o * PIN + r];
    }
}

// ---------------------------------------------------------------------------
// Wc[896 + f*8 + k, o] = sum_d emb[f,k,d] * proj_w[o, 896 + f*128 + d]
__global__ __launch_bounds__(256) void k_build_wc_emb(const float* __restrict__ emb,
                                                      const float* __restrict__ pw,
                                                      float* __restrict__ wc) {
    __shared__ float embf[K_SZ * D_SZ];   // 8x128
    const int f = blockIdx.x;
    const int tid = threadIdx.x;
    #pragma unroll
    for (int i = 0; i < 4; ++i)
        embf[tid + i * 256] = emb[(size_t)f * (K_SZ * D_SZ) + tid + i * 256];
    __syncthreads();

    const int o = tid;  // 0..255
    float acc[K_SZ];
    #pragma unroll
    for (int k = 0; k < K_SZ; ++k) acc[k] = 0.0f;

    const float* wrow = pw + (size_t)o * PIN + TOPK + f * D_SZ;
    #pragma unroll 4
    for (int d = 0; d < D_SZ; ++d) {
        float w = wrow[d];
        #pragma unroll
        for (int k = 0; k < K_SZ; ++k) acc[k] += embf[k * D_SZ + d] * w;
    }
    #pragma unroll
    for (int k = 0; k < K_SZ; ++k)
        wc[(size_t)(TOPK + f * K_SZ + k) * OUT + o] = acc[k];
}

// ---------------------------------------------------------------------------
// A[b, 0:896] = concat(asr[b], mm[b])
__global__ __launch_bounds__(256) void k_copy_embcat(const float* __restrict__ asr,
                                                     const float* __restrict__ mm,
                                                     float* __restrict__ A) {
    int idx = blockIdx.x * blockDim.x + threadIdx.x;
    int n = B_SZ * TOPK;
    for (; idx < n; idx += gridDim.x * blockDim.x) {
        int b = idx / TOPK;
        int c = idx % TOPK;
        float v = (c < ASR) ? asr[b * ASR + c] : mm[b * MM + (c - ASR)];
        A[(size_t)b * KC + c] = v;
    }
}

// ---------------------------------------------------------------------------
// Per-feature tiny MLP + softmax -> p written into A[b, 896 + f*8 + k]
// grid: (F, B/256), block: 256 (thread = one batch row b)
__global__ __launch_bounds__(256) void k_autodis(const float* __restrict__ stat,
                                                 const float* __restrict__ l1w,
                                                 const float* __restrict__ l1b,
                                                 const float* __restrict__ l2w,
                                                 const float* __restrict__ l2b,
                                                 const float* __restrict__ tau,
                                                 float* __restrict__ A) {
    __shared__ float w1s[H_SZ];          // sum over the 3 identical input copies
    __shared__ float b1s[H_SZ];
    __shared__ float w2s[H_SZ * K_SZ];   // [64][8]
    __shared__ float b2s[K_SZ];
    __shared__ float taus[K_SZ];

    const int f = blockIdx.x;
    const int tid = threadIdx.x;
    const int b = blockIdx.y * 256 + tid;

    if (tid < H_SZ) {
        const float* w = l1w + (size_t)f * 3 * H_SZ;
        w1s[tid] = w[tid] + w[H_SZ + tid] + w[2 * H_SZ + tid];
        b1s[tid] = l1b[(size_t)f * H_SZ + tid];
    }
    #pragma unroll
    for (int i = 0; i < 2; ++i)
        w2s[(tid & 255) + i * 256] =
            l2w[(size_t)f * H_SZ * K_SZ + (tid & 255) + i * 256];
    if (tid < K_SZ) {
        b2s[tid]  = l2b[(size_t)f * K_SZ + tid];
        taus[tid] = tau[(size_t)f * K_SZ + tid];
    }
    __syncthreads();

    float x = stat[(size_t)b * F_SZ + f];
    if (x != x || x < 0.0f) x = 0.0f;    // sanitize NaN / negatives

    float h[H_SZ];
    #pragma unroll
    for (int j = 0; j < H_SZ; ++j) {
        float v = x * w1s[j] + b1s[j];
        h[j] = (v >= 0.0f) ? v : 0.01f * v;   // LeakyReLU
    }

    float s[K_SZ];
    #pragma unroll
    for (int k = 0; k < K_SZ; ++k) s[k] = b2s[k];
    #pragma unroll 8
    for (int j = 0; j < H_SZ; ++j) {
        float hv = h[j];
        #pragma unroll
        for (int k = 0; k < K_SZ; ++k) s[k] += hv * w2s[j * K_SZ + k];
    }
    float mx = -3.4e38f;
    #pragma unroll
    for (int k = 0; k < K_SZ; ++k) {
        float v = (s[k] >= 0.0f) ? s[k] : 0.01f * s[k];
        v *= taus[k];
        s[k] = v;
        mx = fmaxf(mx, v);
    }
    float sum = 0.0f;
    #pragma unroll
    for (int k = 0; k < K_SZ; ++k) { s[k] = __expf(s[k] - mx); sum += s[k]; }
    float inv = 1.0f / sum;

    float* dst = A + (size_t)b * KC + TOPK + f * K_SZ;
    #pragma unroll
    for (int k = 0; k < K_SZ; ++k) dst[k] = s[k] * inv;
}

// ---------------------------------------------------------------------------
// out = relu(A[2048 x 4992] @ Wc[4992 x 256] + bias) using v_wmma_f32_16x16x4_f32
// Block: 128 threads (4 waves). Block tile: 64(M) x 64(N). Wave tile: 16 x 64.
// Double-buffered LDS; W tile stored k-pair-interleaved so every B fragment is
// one contiguous 8B ds load (no v_mov repacking before the WMMA).
#define TM 64
#define TN 64
#define TK 32
#define NKB (KC / TK)          // 156
#define ASTRIDE 36             // pad: 16B-aligned b128 stores + conflict-free reads

__global__ __launch_bounds__(128) void k_gemm(const float* __restrict__ A,
                                              const float* __restrict__ W,
                                              const float* __restrict__ bias,
                                              float* __restrict__ out) {
    __shared__ __align__(16) float As[2][TM][ASTRIDE];      // 2 x 64 x 36
    __shared__ __align__(16) float WsP[2][TK / 2][TN][2];   // k-pair interleaved

    const int tid  = threadIdx.x;
    const int wave = tid >> 5;
    const int lane = tid & 31;
    const int row0 = blockIdx.x * TM;
    const int col0 = blockIdx.y * TN;

    v8f acc[4] = {};   // 4 N-tiles of 16x16 per wave

    const int mrow  = wave * 16 + (lane & 15);
    const int khalf = (lane >> 4) * 2;     // lanes 0-15 -> K 0/1, lanes 16-31 -> K 2/3
    const int nlane = lane & 15;

    // ---- tile staging (float4 global loads) ----
    auto stage = [&](int buf, int kb) {
        const int k0 = kb * TK;
        // A tile: 64 rows x 32 k = 512 float4, 4 per thread
        #pragma unroll
        for (int i = 0; i < 4; ++i) {
            int idx = tid + i * 128;
            int r   = idx >> 3;
            int c4  = (idx & 7) * 4;
            float4 v = *(const float4*)(A + (size_t)(row0 + r) * KC + k0 + c4);
            *(float4*)&As[buf][r][c4] = v;
        }
        // W tile: 32 k x 64 n = 512 float4, 4 per thread, scattered k-pair-interleaved
        #pragma unroll
        for (int i = 0; i < 4; ++i) {
            int idx = tid + i * 128;
            int r   = idx >> 4;            // k row 0..31
            int c4  = (idx & 15) * 4;      // n col base
            float4 v = *(const float4*)(W + (size_t)(k0 + r) * OUT + col0 + c4);
            int kp = r >> 1, klo = r & 1;
            WsP[buf][kp][c4 + 0][klo] = v.x;
            WsP[buf][kp][c4 + 1][klo] = v.y;
            WsP[buf][kp][c4 + 2][klo] = v.z;
            WsP[buf][kp][c4 + 3][klo] = v.w;
        }
    };

    stage(0, 0);

    for (int kb = 0; kb < NKB; ++kb) {
        __syncthreads();                   // buf[kb&1] ready; prev reads of buf^1 done
        if (kb + 1 < NKB) stage((kb + 1) & 1, kb + 1);
        if (kb + 2 < NKB) {                // gfx1250 global_prefetch_b8, 2 blocks ahead
            __builtin_prefetch(W + (size_t)((kb + 2) * TK + (tid >> 2)) * OUT
                                 + col0 + (tid & 3) * 16, 0, 3);
        }

        const int cur = kb & 1;
        #pragma unroll
        for (int kk = 0; kk < 8; ++kk) {           // 8 k-steps of 4
            v2f a = *(const v2f*)&As[cur][mrow][kk * 4 + khalf];
            const int kp = kk * 2 + (lane >> 4);   // (kk*4 + khalf) / 2
            #pragma unroll
            for (int nt = 0; nt < 4; ++nt) {
                v2f bfrag = *(const v2f*)&WsP[cur][kp][nt * 16 + nlane][0];
                acc[nt] = __builtin_amdgcn_wmma_f32_16x16x4_f32(
                    false, a, false, bfrag, (short)0, acc[nt], false, false);
            }
        }
    }

    // Epilogue: C layout -> VGPR r: lanes 0-15 = row r, lanes 16-31 = row r+8
    const int mbase = row0 + wave * 16 + ((lane >> 4) ? 8 : 0);
    #pragma unroll
    for (int nt = 0; nt < 4; ++nt) {
        int col = col0 + nt * 16 + nlane;
        float bv = bias[col];
        #pragma unroll
        for (int r = 0; r < 8; ++r) {
            float v = acc[nt][r] + bv;
            out[(size_t)(mbase + r) * OUT + col] = (v > 0.0f) ? v : 0.0f;
        }
    }
}

// ---------------------------------------------------------------------------
extern "C" void kernel_launch(void* const* d_in, const int* in_sizes, int n_in,
                              void* d_out, int out_size, void* d_ws, size_t ws_size,
                              hipStream_t stream) {
    const float* stat = (const float*)d_in[0];   // [2048,512]
    const float* asr  = (const float*)d_in[1];   // [2048,128]
    const float* mm   = (const float*)d_in[2];   // [2048,768]
    const float* l1w  = (const float*)d_in[3];   // [512,3,64]
    const float* l1b  = (const float*)d_in[4];   // [512,1,64]
    const float* l2w  = (const float*)d_in[5];   // [512,64,8]
    const float* l2b  = (const float*)d_in[6];   // [512,1,8]
    const float* tau  = (const float*)d_in[7];   // [512,1,8]
    const float* emb  = (const float*)d_in[8];   // [512,8,128]
    const float* pw   = (const float*)d_in[9];   // [256,66432]
    const float* pb   = (const float*)d_in[10];  // [256]
    float* out = (float*)d_out;                  // [2048,256]

    // workspace layout: A [2048 x 4992] | Wc [4992 x 256]
    float* A  = (float*)d_ws;
    float* Wc = A + (size_t)B_SZ * KC;

    // 1) Wc top rows (transpose of proj_w asr/mm columns)
    k_build_wc_top<<<dim3(224), dim3(256), 0, stream>>>(pw, Wc);
    // 2) Wc folded-emb rows: M[f] = emb[f] @ proj_w_slice^T
    k_build_wc_emb<<<dim3(F_SZ), dim3(256), 0, stream>>>(emb, pw, Wc);
    // 3) A[:, 0:896] = [asr | mm]
    k_copy_embcat<<<dim3(896), dim3(256), 0, stream>>>(asr, mm, A);
    // 4) A[:, 896:] = softmax probabilities p[b,f,k]
    k_autodis<<<dim3(F_SZ, B_SZ / 256), dim3(256), 0, stream>>>(
        stat, l1w, l1b, l2w, l2b, tau, A);
    // 5) fused GEMM + bias + relu (fp32 WMMA, double-buffered LDS)
    k_gemm<<<dim3(B_SZ / TM, OUT / TN), dim3(128), 0, stream>>>(A, Wc, pb, out);
}